// KNeighborsModel_62294205661725
// MI455X (gfx1250) — compile-verified
//
#include <hip/hip_runtime.h>
#include <hip/hip_bf16.h>
#include <math.h>

typedef __attribute__((ext_vector_type(2))) float v2f;
typedef __attribute__((ext_vector_type(8))) float v8f;

#define DIM      128   // feature dim
#define KCH      32    // 128 / 4 (K per WMMA_F32_16X16X4_F32)
#define NSPLIT   16    // blocks splitting the N (train) dimension
#define TILE_N   16    // train columns per WMMA tile
#define ROWS_PB  64    // query rows per block (4 waves x 16)
#define BSTRIDE  132   // 128 + 4 pad floats -> conflict-free frag reads, 16B-aligned rows

struct Pair { float v; int i; };

// ---------------------------------------------------------------------------
// t2[n] = sum_d train[n][d]^2   (one wave32 per row, float4 per lane)
// ---------------------------------------------------------------------------
__global__ void knn_row_norms(const float* __restrict__ td, float* __restrict__ t2, int N) {
  const int row  = blockIdx.x * 8 + (threadIdx.x >> 5);
  const int lane = threadIdx.x & 31;
  if (row >= N) return;
  float4 v = ((const float4*)(td + (size_t)row * DIM))[lane];
  float s = v.x * v.x + v.y * v.y + v.z * v.z + v.w * v.w;
  #pragma unroll
  for (int off = 16; off > 0; off >>= 1) s += __shfl_xor(s, off, 32);
  if (lane == 0) t2[row] = s;
}

// ---------------------------------------------------------------------------
// Fused (-2 x.y^T + ||y||^2 + ||x||^2) + streaming top-5, per N-split.
// Block: 128 threads = 4 waves; wave w owns query rows [blk*64+16w, +16).
// ---------------------------------------------------------------------------
__global__ __launch_bounds__(128)
void knn_main(const float* __restrict__ x, const float* __restrict__ td,
              const float* __restrict__ t2, Pair* __restrict__ partial,
              int N) {
  // LDS: staging buffer [16][132] + x2[4][16] during scan; candidate lists after.
  __shared__ __align__(16) char smem[64 * 16 * 5 * 8];     // 40960 B
  float* bt   = (float*)smem;                              // [16][BSTRIDE]
  float* x2s  = (float*)(smem + 16 * BSTRIDE * 4);         // [4][16]
  float* cval = (float*)smem;                              // [64][16][5]
  int*   cidx = (int*)(smem + 64 * 16 * 5 * 4);            // [64][16][5]

  const int tid  = threadIdx.x;
  const int wave = tid >> 5;
  const int lane = tid & 31;
  const int nl   = lane & 15;   // N-column within tile / M-row of A frag
  const int half = lane >> 4;   // K phase (A/B frags), M phase (C/D)

  // ---- A fragments: queries scaled by -2 (held in registers all kernel) ----
  const int    mrow = blockIdx.x * ROWS_PB + wave * 16 + nl;
  const float* xrow = x + (size_t)mrow * DIM;
  v2f   afrag[KCH];
  float p = 0.f;
  #pragma unroll
  for (int j = 0; j < KCH; ++j) {
    float a0 = xrow[4 * j + 2 * half];
    float a1 = xrow[4 * j + 2 * half + 1];
    afrag[j].x = -2.f * a0;
    afrag[j].y = -2.f * a1;
    p += a0 * a0 + a1 * a1;
  }
  p += __shfl_xor(p, 16, 32);              // combine the two K-halves of row nl
  if (half == 0) x2s[wave * 16 + nl] = p;  // ||x_row||^2
  __syncthreads();
  float x2v[8];
  #pragma unroll
  for (int r = 0; r < 8; ++r) x2v[r] = x2s[wave * 16 + 8 * half + r];

  // ---- per-lane top-5 (sorted ascending) for this lane's 8 C/D rows ----
  float tv[8][5];
  int   ti[8][5];
  #pragma unroll
  for (int r = 0; r < 8; ++r)
    #pragma unroll
    for (int k = 0; k < 5; ++k) { tv[r][k] = __builtin_inff(); ti[r][k] = 0; }

  const int perSplit = (N + NSPLIT - 1) / NSPLIT;
  const int nBeg     = blockIdx.y * perSplit;
  const int nEnd     = min(N, nBeg + perSplit);
  const int nTiles   = (nEnd - nBeg + TILE_N - 1) / TILE_N;

  for (int t = 0; t < nTiles; ++t) {
    const int n0 = nBeg + t * TILE_N;
    __syncthreads();                        // previous tile's frag reads done
    {   // cooperative stage of 16x128 train tile, row-major + pad
      const int    r   = tid >> 3;          // 0..15 tile row
      const int    c   = (tid & 7) * 16;    // 0..112 col base
      const int    n   = n0 + r;
      const float* src = td + (size_t)min(n, N - 1) * DIM + c;
      const bool   ok  = (n < nEnd);
      #pragma unroll
      for (int q = 0; q < 4; ++q) {
        float4 vv = ((const float4*)src)[q];          // address clamped: safe
        if (!ok) vv = make_float4(0.f, 0.f, 0.f, 0.f);
        *(float4*)&bt[r * BSTRIDE + c + 4 * q] = vv;
      }
      if (t + 1 < nTiles)                    // WGP-scope prefetch of next tile
        __builtin_prefetch(src + TILE_N * DIM, 0, 3);
    }
    __syncthreads();

    const int   n   = n0 + nl;
    const float t2v = (n < nEnd) ? t2[min(n, N - 1)] : __builtin_inff();

    // two independent accumulation chains -> ILP across the XDL pipe
    v8f acc0 = {};
    v8f acc1 = {};
    const float* bb = &bt[nl * BSTRIDE + 2 * half];
    #pragma unroll
    for (int j = 0; j < KCH; j += 2) {
      v2f b0, b1;
      b0.x = bb[4 * j];
      b0.y = bb[4 * j + 1];
      b1.x = bb[4 * j + 4];
      b1.y = bb[4 * j + 5];
      acc0 = __builtin_amdgcn_wmma_f32_16x16x4_f32(
          false, afrag[j], false, b0, (short)0, acc0, false, false);
      acc1 = __builtin_amdgcn_wmma_f32_16x16x4_f32(
          false, afrag[j + 1], false, b1, (short)0, acc1, false, false);
    }

    #pragma unroll
    for (int r = 0; r < 8; ++r) {            // d2 = ||x||^2 - 2x.y + ||y||^2
      const float d2 = (x2v[r] + t2v) + (acc0[r] + acc1[r]);
      if (d2 < tv[r][4]) {                   // rare after warm-up
        tv[r][4] = d2; ti[r][4] = n;
        #pragma unroll
        for (int s = 4; s >= 1; --s) {
          if (tv[r][s] < tv[r][s - 1]) {
            float tf = tv[r][s]; tv[r][s] = tv[r][s - 1]; tv[r][s - 1] = tf;
            int   tn = ti[r][s]; ti[r][s] = ti[r][s - 1]; ti[r][s - 1] = tn;
          }
        }
      }
    }
  }

  // ---- block-level reduction of per-lane lists -> per-row top-5 ----
  __syncthreads();
  #pragma unroll
  for (int r = 0; r < 8; ++r) {
    const int mloc = 8 * half + r;
    const int base = ((wave * 16 + mloc) * 16 + nl) * 5;
    #pragma unroll
    for (int k = 0; k < 5; ++k) { cval[base + k] = tv[r][k]; cidx[base + k] = ti[r][k]; }
  }
  __syncthreads();
  if (tid < ROWS_PB) {                       // one thread per block-local row
    float v[5]; int id[5];
    #pragma unroll
    for (int k = 0; k < 5; ++k) { v[k] = __builtin_inff(); id[k] = 0; }
    const int base = tid * 16 * 5;
    for (int s = 0; s < 16 * 5; ++s) {
      const float c = cval[base + s];
      if (c < v[4]) {
        v[4] = c; id[4] = cidx[base + s];
        #pragma unroll
        for (int q = 4; q >= 1; --q)
          if (v[q] < v[q - 1]) {
            float tf = v[q]; v[q] = v[q - 1]; v[q - 1] = tf;
            int   tn = id[q]; id[q] = id[q - 1]; id[q - 1] = tn;
          }
      }
    }
    const int b = blockIdx.x * ROWS_PB + tid;
    Pair* dst = partial + ((size_t)b * NSPLIT + blockIdx.y) * 5;
    #pragma unroll
    for (int k = 0; k < 5; ++k) { dst[k].v = v[k]; dst[k].i = id[k]; }
  }
}

// ---------------------------------------------------------------------------
// Merge NSPLIT partial top-5s per query; inverse-distance weighted labels.
// ---------------------------------------------------------------------------
__global__ void knn_final(const Pair* __restrict__ partial,
                          const float* __restrict__ labels,
                          float* __restrict__ out, int B) {
  const int b = blockIdx.x * blockDim.x + threadIdx.x;
  if (b >= B) return;
  float v[5]; int id[5];
  #pragma unroll
  for (int k = 0; k < 5; ++k) { v[k] = __builtin_inff(); id[k] = 0; }
  const Pair* p = partial + (size_t)b * (NSPLIT * 5);
  for (int s = 0; s < NSPLIT * 5; ++s) {
    const Pair c = p[s];
    if (c.v < v[4]) {
      v[4] = c.v; id[4] = c.i;
      #pragma unroll
      for (int q = 4; q >= 1; --q)
        if (v[q] < v[q - 1]) {
          float tf = v[q]; v[q] = v[q - 1]; v[q - 1] = tf;
          int   tn = id[q]; id[q] = id[q - 1]; id[q - 1] = tn;
        }
    }
  }
  float d[5]; bool exact = false;
  #pragma unroll
  for (int k = 0; k < 5; ++k) {
    d[k] = sqrtf(fmaxf(v[k], 0.f));
    exact |= (d[k] == 0.f);
  }
  float wsum = 0.f, acc = 0.f;
  #pragma unroll
  for (int k = 0; k < 5; ++k) {
    const float w   = exact ? (d[k] == 0.f ? 1.f : 0.f) : 1.f / d[k];
    const float lab = labels[id[k] < 0 ? 0 : id[k]];
    wsum += w; acc += w * lab;
  }
  out[b] = acc / wsum;
}

// ---------------------------------------------------------------------------
extern "C" void kernel_launch(void* const* d_in, const int* in_sizes, int n_in,
                              void* d_out, int out_size, void* d_ws, size_t ws_size,
                              hipStream_t stream) {
  const float* x  = (const float*)d_in[0];   // [B,128]
  const float* td = (const float*)d_in[1];   // [N,128]
  const float* tl = (const float*)d_in[2];   // [N]
  const int B = in_sizes[0] / DIM;           // 2048
  const int N = in_sizes[2];                 // 100000

  // workspace: t2[N] floats, then Pair partial[B][NSPLIT][5]
  float* t2 = (float*)d_ws;
  size_t off = ((size_t)N * sizeof(float) + 255) & ~(size_t)255;
  Pair* partial = (Pair*)((char*)d_ws + off);

  knn_row_norms<<<(N + 7) / 8, 256, 0, stream>>>(td, t2, N);

  dim3 grid(B / ROWS_PB, NSPLIT);
  knn_main<<<grid, 128, 0, stream>>>(x, td, t2, partial, N);

  knn_final<<<(B + 255) / 256, 256, 0, stream>>>(partial, tl, (float*)d_out, B);
}